// DiscriminativeLoss_9380208575089
// MI455X (gfx1250) — compile-verified
//
#include <hip/hip_runtime.h>

#define NB   32
#define NF   8
#define NPT  131072
#define NC   16
#define D_VAR 0.5f
#define D_DIST 1.5f
#define GAMMA 0.001f

#define BX    32      // blocks per batch for heavy kernels
#define WAVES 8       // waves per block (256 threads, wave32)
#define TILE  128     // points staged per wave per tile

typedef __attribute__((ext_vector_type(2))) float v2f;
typedef __attribute__((ext_vector_type(8))) float v8f;

// ---------------- kernel 0: zero workspace accumulators ----------------
__global__ void k_zero(float* ws, int n) {
    int i = blockIdx.x * blockDim.x + threadIdx.x;
    if (i < n) ws[i] = 0.f;
}

// ---------------- kernel 1: segment sums + counts via WMMA -------------
// segSums layout: [NB][NC][9]  (cols 0..7 = feature sums, col 8 = count)
__global__ __launch_bounds__(256) void k_segsum(const float* __restrict__ x,
                                                const int* __restrict__ tgt,
                                                float* __restrict__ segSums) {
    __shared__ float xbuf[WAVES][NF][TILE + 4];   // stride 132: conflict-friendly
    __shared__ int   lbuf[WAVES][TILE];
    __shared__ float acc[NC][9];

    const int tid  = threadIdx.x;
    const int wv   = tid >> 5;
    const int lane = tid & 31;
    const int b    = blockIdx.y;

    if (tid < NC * 9) acc[tid / 9][tid % 9] = 0.f;
    __syncthreads();

    const int m     = lane & 15;            // A-matrix row (cluster id)
    const int khalf = (lane >> 4) * 2;      // 0 for lanes 0-15, 2 for 16-31
    const int col   = lane & 15;            // B/C column

    const size_t xbase = (size_t)b * NF * (size_t)NPT;
    const int pointsPerWave = NPT / (BX * WAVES);   // 512
    const int n0w = (blockIdx.x * WAVES + wv) * pointsPerWave;

    v8f c = {};  // 16x16 f32 accumulator (striped across wave)

    for (int t = 0; t < pointsPerWave; t += TILE) {
        const int n0 = n0w + t;
        // coalesced stage: 8 features x 128 points + 128 labels
        #pragma unroll
        for (int f = 0; f < NF; ++f) {
            #pragma unroll
            for (int i = 0; i < TILE / 32; ++i)
                xbuf[wv][f][i * 32 + lane] =
                    x[xbase + (size_t)f * NPT + n0 + i * 32 + lane];
        }
        #pragma unroll
        for (int i = 0; i < TILE / 32; ++i)
            lbuf[wv][i * 32 + lane] = tgt[(size_t)b * NPT + n0 + i * 32 + lane];

        // 32 WMMA steps of 4 points each (same-wave LDS, in-order, no barrier)
        #pragma unroll 4
        for (int s = 0; s < TILE; s += 4) {
            const int p = s + khalf;
            const int l0 = lbuf[wv][p];
            const int l1 = lbuf[wv][p + 1];
            v2f a, bb;
            a.x = (l0 == m) ? 1.f : 0.f;     // one-hot A: 16 clusters x 4 pts
            a.y = (l1 == m) ? 1.f : 0.f;
            if (col < NF) {                  // B: 4 pts x (8 feat | count | 0)
                bb.x = xbuf[wv][col][p];
                bb.y = xbuf[wv][col][p + 1];
            } else if (col == 8) {
                bb.x = 1.f; bb.y = 1.f;
            } else {
                bb.x = 0.f; bb.y = 0.f;
            }
            c = __builtin_amdgcn_wmma_f32_16x16x4_f32(
                    false, a, false, bb, (short)0, c, false, false);
        }
    }

    // C layout: VGPR r, lanes 0-15 -> row r, lanes 16-31 -> row r+8; col = lane&15
    const int rowBase = (lane >> 4) * 8;
    if (col < 9) {
        #pragma unroll
        for (int r = 0; r < 8; ++r)
            atomicAdd(&acc[rowBase + r][col], c[r]);
    }
    __syncthreads();
    if (tid < NC * 9)
        atomicAdd(&segSums[b * NC * 9 + tid], acc[tid / 9][tid % 9]);
}

// ---------------- kernel 2: variance term (second streaming pass) -------
__global__ __launch_bounds__(256) void k_var(const float* __restrict__ x,
                                             const int* __restrict__ tgt,
                                             const float* __restrict__ segSums,
                                             float* __restrict__ varAcc) {
    __shared__ float sRaw[NC * 9];
    __shared__ float sMean[NC][NF];
    __shared__ float sInv[NC];
    __shared__ float sSum;

    const int tid = threadIdx.x;
    const int b   = blockIdx.y;

    if (tid < NC * 9) sRaw[tid] = segSums[b * NC * 9 + tid];
    if (tid == 0) sSum = 0.f;
    __syncthreads();
    if (tid < NC) {
        float cnt = sRaw[tid * 9 + 8];
        sInv[tid] = (cnt > 0.f) ? 1.f / cnt : 0.f;
    }
    if (tid < NC * NF) {
        int cc = tid >> 3, f = tid & 7;
        float cnt = sRaw[cc * 9 + 8];
        sMean[cc][f] = sRaw[cc * 9 + f] / ((cnt > 0.f) ? cnt : 1.f);
    }
    __syncthreads();

    const size_t xbase = (size_t)b * NF * (size_t)NPT;
    float local = 0.f;
    for (int n = blockIdx.x * blockDim.x + tid; n < NPT;
         n += gridDim.x * blockDim.x) {
        const int l = tgt[(size_t)b * NPT + n];
        float s = 0.f;
        #pragma unroll
        for (int f = 0; f < NF; ++f)
            s += fabsf(x[xbase + (size_t)f * NPT + n] - sMean[l][f]);
        float h = s - D_VAR;
        h = (h > 0.f) ? h : 0.f;
        local += h * h * sInv[l];   // already folds the /count
    }
    atomicAdd(&sSum, local);
    __syncthreads();
    if (tid == 0) atomicAdd(varAcc, sSum);
}

// ---------------- kernel 3: dist + reg terms, finalize ------------------
__global__ __launch_bounds__(256) void k_final(const int* __restrict__ tgt,
                                               const float* __restrict__ segSums,
                                               const float* __restrict__ varAcc,
                                               float* __restrict__ out) {
    __shared__ float sD, sR;
    const int tid = threadIdx.x;
    if (tid == 0) { sD = 0.f; sR = 0.f; }
    __syncthreads();

    float d_local = 0.f, r_local = 0.f;

    // dist: bug-compatible — means of clusters of the FIRST 16 POINTS per batch
    for (int idx = tid; idx < NB * NC * NC; idx += blockDim.x) {
        const int b = idx >> 8;
        const int i = (idx >> 4) & 15;
        const int j = idx & 15;
        const int li = tgt[(size_t)b * NPT + i];
        const int lj = tgt[(size_t)b * NPT + j];
        const float* si = &segSums[(b * NC + li) * 9];
        const float* sj = &segSums[(b * NC + lj) * 9];
        float ci = si[8]; ci = (ci > 0.f) ? ci : 1.f;
        float cj = sj[8]; cj = (cj > 0.f) ? cj : 1.f;
        float d = 0.f;
        #pragma unroll
        for (int f = 0; f < NF; ++f)
            d += fabsf(si[f] / ci - sj[f] / cj);
        const float margin = (i == j) ? 0.f : 2.f * D_DIST;
        float h = margin - d;
        h = (h > 0.f) ? h : 0.f;
        d_local += h * h / (float)(NC * (NC - 1));
    }

    // reg: mean over clusters of L1(mean)
    for (int idx = tid; idx < NB * NC; idx += blockDim.x) {
        const float* s = &segSums[idx * 9];
        float cnt = s[8]; cnt = (cnt > 0.f) ? cnt : 1.f;
        float l1 = 0.f;
        #pragma unroll
        for (int f = 0; f < NF; ++f) l1 += fabsf(s[f] / cnt);
        r_local += l1 / (float)NC;
    }

    atomicAdd(&sD, d_local);
    atomicAdd(&sR, r_local);
    __syncthreads();
    if (tid == 0)
        out[0] = (varAcc[0] + sD + GAMMA * sR) / (float)NB;
}

// ------------------------------ launch ----------------------------------
extern "C" void kernel_launch(void* const* d_in, const int* in_sizes, int n_in,
                              void* d_out, int out_size, void* d_ws, size_t ws_size,
                              hipStream_t stream) {
    (void)in_sizes; (void)n_in; (void)out_size; (void)ws_size;
    const float* x   = (const float*)d_in[0];
    const int*   tgt = (const int*)d_in[1];
    float* out = (float*)d_out;
    float* ws  = (float*)d_ws;

    float* segSums = ws;                 // NB*NC*9 = 4608 floats
    float* varAcc  = ws + NB * NC * 9;   // 1 float
    const int nzero = NB * NC * 9 + 1;

    k_zero  <<<dim3((nzero + 255) / 256), 256, 0, stream>>>(ws, nzero);
    k_segsum<<<dim3(BX, NB), 256, 0, stream>>>(x, tgt, segSums);
    k_var   <<<dim3(32, NB), 256, 0, stream>>>(x, tgt, segSums, varAcc);
    k_final <<<dim3(1), 256, 0, stream>>>(tgt, segSums, varAcc, out);
}